// GIN_1709396984307
// MI455X (gfx1250) — compile-verified
//
#include <hip/hip_runtime.h>
#include <cstdint>

#define C_DIM   128
#define LDS_PAD 132   // row stride in floats; 132 % 64 == 4 -> spreads LDS banks

typedef float v2f __attribute__((ext_vector_type(2)));
typedef float v8f __attribute__((ext_vector_type(8)));

// ---------------------------------------------------------------------------
// Guaranteed hardware fp32 atomic add (no-return form, device scope).
// Avoids any chance of LLVM expanding atomicrmw fadd into a CAS loop:
// emits exactly one GLOBAL_ATOMIC_ADD_F32 (STOREcnt-tracked).
// ---------------------------------------------------------------------------
__device__ __forceinline__ void atomic_add_f32_dev(float* p, float v) {
    asm volatile("global_atomic_add_f32 %0, %1, off scope:SCOPE_DEV"
                 :
                 : "v"((uint64_t)(uintptr_t)p), "v"(v)
                 : "memory");
}

// ---------------------------------------------------------------------------
// zero workspace accumulator
// ---------------------------------------------------------------------------
__global__ void gine_zero_f4(float4* __restrict__ p, int n4) {
    int i = blockIdx.x * blockDim.x + threadIdx.x;
    if (i < n4) p[i] = make_float4(0.f, 0.f, 0.f, 0.f);
}

// ---------------------------------------------------------------------------
// Edge phase: m_e = relu(x[src] + edge_attr*W_e + b_e); aggr[dst] += m_e
// One wave32 per edge; lane owns 4 channels (float4) -> coalesced 512B row.
// Node table (5.1 MB) and accumulator live in the 192 MB L2, so both the
// random gather and the atomic scatter resolve in L2.
// ---------------------------------------------------------------------------
__global__ void gine_edge_scatter(const float* __restrict__ xin,
                                  const int*   __restrict__ ei,   // [2,E]
                                  const float* __restrict__ ea,   // [E]
                                  const float* __restrict__ We,   // [C]
                                  const float* __restrict__ be,   // [C]
                                  float*       __restrict__ aggr, // [N,C]
                                  int E) {
    const int e    = (blockIdx.x * blockDim.x + threadIdx.x) >> 5; // wave id
    const int lane = threadIdx.x & 31;
    if (e >= E) return;

    const int   s = ei[e];
    const int   d = ei[E + e];
    const float a = ea[e];

    const int c0 = lane * 4;
    const float4 xv = *(const float4*)(xin + (size_t)s * C_DIM + c0);
    const float4 wv = *(const float4*)(We + c0);
    const float4 bv = *(const float4*)(be + c0);

    float4 m;
    m.x = fmaxf(fmaf(a, wv.x, bv.x) + xv.x, 0.f);
    m.y = fmaxf(fmaf(a, wv.y, bv.y) + xv.y, 0.f);
    m.z = fmaxf(fmaf(a, wv.z, bv.z) + xv.z, 0.f);
    m.w = fmaxf(fmaf(a, wv.w, bv.w) + xv.w, 0.f);

    float* o = aggr + (size_t)d * C_DIM + c0;
    atomic_add_f32_dev(o + 0, m.x);
    atomic_add_f32_dev(o + 1, m.y);
    atomic_add_f32_dev(o + 2, m.z);
    atomic_add_f32_dev(o + 3, m.w);
}

// ---------------------------------------------------------------------------
// Node phase: out = (aggr + x) @ W.T + b  [+relu], via V_WMMA_F32_16X16X4_F32.
// Block = 256 threads = 8 waves. Block owns a 16-row M-tile; wave w owns
// output columns [16w, 16w+16). K-loop: 32 WMMAs of 16x16x4 f32.
//
// ISA VGPR layouts (wave32):
//   A 16x4 f32 : lane L -> row M=L&15; VGPR0={K0|K2}, VGPR1={K1|K3} split on L>>4
//   B 4x16 f32 : lane L -> col N=L&15; same K split
//   C/D 16x16  : lane L -> col N=L&15; VGPR v -> row M = v + 8*(L>>4)
// ---------------------------------------------------------------------------
__global__ void gine_node_gemm(const float* __restrict__ aggr,  // [N,C]
                               const float* __restrict__ xin,   // [N,C]
                               const float* __restrict__ W,     // [C,C] row-major (out ch major)
                               const float* __restrict__ bias,  // [C]
                               float*       __restrict__ out,   // [N,C]
                               int Nnodes, int relu_out) {
    __shared__ float s_in[16 * LDS_PAD];

    const int tid  = threadIdx.x;
    const int tile = blockIdx.x;

    // stage (aggr + x) 16x128 tile into LDS: 512 float4s across 256 threads
    #pragma unroll
    for (int i = 0; i < 2; ++i) {
        const int idx = tid + i * 256;        // 0..511
        const int m   = idx >> 5;             // 32 float4 per row
        const int c4  = (idx & 31) * 4;
        int g = tile * 16 + m;
        if (g >= Nnodes) g = Nnodes - 1;      // clamp (uniform-per-block safety)
        const size_t off = (size_t)g * C_DIM + c4;
        const float4 av = *(const float4*)(aggr + off);
        const float4 xv = *(const float4*)(xin + off);
        float4 sv;
        sv.x = av.x + xv.x; sv.y = av.y + xv.y;
        sv.z = av.z + xv.z; sv.w = av.w + xv.w;
        *(float4*)(&s_in[m * LDS_PAD + c4]) = sv;
    }
    __syncthreads();

    const int wave = tid >> 5;
    const int lane = tid & 31;
    const int mrow = lane & 15;               // A row
    const int ncol = wave * 16 + (lane & 15); // B/output column
    const int ko   = (lane >> 4) * 2;         // K sub-offset per lane half

    const float* wrow = W + (size_t)ncol * C_DIM;   // W[n][k]
    const float* arow = s_in + mrow * LDS_PAD;

    v8f acc = {};
    #pragma unroll
    for (int kb = 0; kb < C_DIM; kb += 4) {
        v2f av, bv;
        av.x = arow[kb + ko];
        av.y = arow[kb + ko + 1];
        bv.x = wrow[kb + ko];
        bv.y = wrow[kb + ko + 1];
        acc = __builtin_amdgcn_wmma_f32_16x16x4_f32(
            /*neg_a=*/false, av, /*neg_b=*/false, bv,
            /*c_mod=*/(short)0, acc, /*reuse_a=*/false, /*reuse_b=*/false);
    }

    const float bn = bias[ncol];
    const int   mo = (lane >> 4) * 8;
    #pragma unroll
    for (int v = 0; v < 8; ++v) {
        const int g = tile * 16 + v + mo;
        float val = acc[v] + bn;
        if (relu_out) val = fmaxf(val, 0.f);
        if (g < Nnodes) out[(size_t)g * C_DIM + ncol] = val;
    }
}

// ---------------------------------------------------------------------------
// launch: zero -> edge -> gemm (relu)  |  zero -> edge -> gemm
// ---------------------------------------------------------------------------
extern "C" void kernel_launch(void* const* d_in, const int* in_sizes, int n_in,
                              void* d_out, int out_size, void* d_ws, size_t ws_size,
                              hipStream_t stream) {
    const float* x   = (const float*)d_in[0];
    const int*   ei  = (const int*)  d_in[1];
    const float* ea  = (const float*)d_in[2];
    const float* We1 = (const float*)d_in[3];
    const float* be1 = (const float*)d_in[4];
    const float* W1  = (const float*)d_in[5];
    const float* b1  = (const float*)d_in[6];
    const float* We2 = (const float*)d_in[7];
    const float* be2 = (const float*)d_in[8];
    const float* W2  = (const float*)d_in[9];
    const float* b2  = (const float*)d_in[10];
    float*       out = (float*)d_out;

    const int NC = in_sizes[0];          // N*C
    const int N  = NC / C_DIM;           // 10000
    const int E  = in_sizes[2];          // 640000

    float* aggr = (float*)d_ws;          // [N,C], zeroed per layer
    float* h    = aggr + NC;             // [N,C] hidden activations

    const int n4        = NC / 4;
    const int zeroGrid  = (n4 + 255) / 256;
    const int edgeGrid  = (E + 7) / 8;           // 8 edges (waves) per 256-thr block
    const int gemmGrid  = (N + 15) / 16;         // 16-row M-tile per block

    // ---- layer 1 ----
    gine_zero_f4<<<zeroGrid, 256, 0, stream>>>((float4*)aggr, n4);
    gine_edge_scatter<<<edgeGrid, 256, 0, stream>>>(x, ei, ea, We1, be1, aggr, E);
    gine_node_gemm<<<gemmGrid, 256, 0, stream>>>(aggr, x, W1, b1, h, N, /*relu=*/1);

    // ---- layer 2 ----
    gine_zero_f4<<<zeroGrid, 256, 0, stream>>>((float4*)aggr, n4);
    gine_edge_scatter<<<edgeGrid, 256, 0, stream>>>(h, ei, ea, We2, be2, aggr, E);
    gine_node_gemm<<<gemmGrid, 256, 0, stream>>>(aggr, h, W2, b2, out, N, /*relu=*/0);
}